// RandSpanNNShot_82892868812869
// MI455X (gfx1250) — compile-verified
//
#include <hip/hip_runtime.h>

typedef __attribute__((ext_vector_type(2))) float v2f;
typedef __attribute__((ext_vector_type(8))) float v8f;

#define B_     512
#define L_     128
#define H_     768
#define NSPAN_ 4096
#define NCLS_  6
#define D_     1536   // 2*H

#define TM 128        // block tile M (queries)
#define TN 64         // block tile N (supports)
#define KC 16         // K chunk per LDS stage
#define LDA (KC + 1)  // 17
#define LDB (KC + 1)  // 17
#define LDD (TN + 1)  // 65

// ---------------- float atomic max via CAS (robust, compiles everywhere) ----
__device__ inline void atomicMaxF(float* addr, float val) {
  unsigned int* ia = (unsigned int*)addr;
  unsigned int old = *ia;
  while (__uint_as_float(old) < val) {
    unsigned int assumed = old;
    old = atomicCAS(ia, assumed, __float_as_uint(val));
    if (old == assumed) break;
  }
}

// ---------------- order = argsort(1 - mask, stable) per row -----------------
__global__ void order_kernel(const int* __restrict__ mask,
                             int* __restrict__ order) {
  __shared__ int m[L_];
  int row = blockIdx.x;
  int i = threadIdx.x;          // blockDim.x == 128
  m[i] = mask[row * L_ + i];
  __syncthreads();
  int ones_before = 0;
  for (int j = 0; j < i; ++j) ones_before += m[j];
  int total = ones_before;
  for (int j = i; j < L_; ++j) total += m[j];
  int pos = m[i] ? ones_before : (total + (i - ones_before));
  order[row * L_ + pos] = i;
}

// ---------------- gather span vectors (4096 x 1536) + squared norms ---------
__global__ void gather_kernel(const float* __restrict__ emb,
                              const int* __restrict__ span_id,
                              const int* __restrict__ order,
                              float* __restrict__ spans,
                              float* __restrict__ norms) {
  __shared__ float red[256];
  int s = blockIdx.x;
  int tid = threadIdx.x;        // 256 threads
  int sent  = span_id[s * 3 + 0];
  int start = order[sent * L_ + span_id[s * 3 + 1]];
  int end   = order[sent * L_ + span_id[s * 3 + 2]];
  const float4* rs = (const float4*)(emb + ((size_t)sent * L_ + start) * H_);
  const float4* re = (const float4*)(emb + ((size_t)sent * L_ + end)   * H_);
  float4* dst = (float4*)(spans + (size_t)s * D_);
  float acc = 0.f;
  for (int j = tid; j < (D_ / 4); j += 256) {   // 384 float4
    float4 v = (j < (H_ / 4)) ? rs[j] : re[j - (H_ / 4)];
    dst[j] = v;
    acc += v.x * v.x + v.y * v.y + v.z * v.z + v.w * v.w;
  }
  red[tid] = acc;
  __syncthreads();
  for (int off = 128; off > 0; off >>= 1) {
    if (tid < off) red[tid] += red[tid + off];
    __syncthreads();
  }
  if (tid == 0) norms[s] = red[0];
}

// ---------------- init logits to -inf ---------------------------------------
__global__ void init_logits(float* __restrict__ logits, int n) {
  int i = blockIdx.x * blockDim.x + threadIdx.x;
  if (i < n) logits[i] = -__builtin_inff();
}

#define WMMA_F32(a, b, c)                                              \
  __builtin_amdgcn_wmma_f32_16x16x4_f32(false, (a), false, (b),        \
                                        (short)0, (c), false, false)

// ---------------- WMMA GEMM (qs = Q . S^T) fused with per-class max ---------
// 8 waves, each owns a 32x32 tile (acc[2][2]) for 4x fragment reuse.
// Double-buffered LDS K-stages with register prefetch.
__global__ __launch_bounds__(256)
void nn_gemm_kernel(const float* __restrict__ Q,      // [NSPAN][D]
                    const float* __restrict__ S,      // [NSPAN][D]
                    const float* __restrict__ q2,     // [NSPAN]
                    const float* __restrict__ s2,     // [NSPAN]
                    const int*   __restrict__ label,  // [NSPAN]
                    float* __restrict__ logits) {     // [NSPAN][NCLS], -inf init
  __shared__ float As[2][TM * LDA];
  __shared__ float Bs[2][TN * LDB];
  __shared__ float Ds[TM * LDD];
  __shared__ int   slab[TN];
  __shared__ float s2t[TN];

  int tid  = threadIdx.x;
  int wave = tid >> 5;           // 0..7
  int lane = tid & 31;
  int wm   = wave >> 1;          // 0..3 : 32-row group within 128
  int wn   = wave & 1;           // 0..1 : 32-col group within 64
  int hi   = lane >> 4;          // 0/1
  int lo   = lane & 15;

  int n0 = blockIdx.x * TN;      // support tile base
  int m0 = blockIdx.y * TM;      // query tile base

  if (tid < TN) {
    slab[tid] = label[n0 + tid];
    s2t[tid]  = s2[n0 + tid];
  }

  // staging task decomposition: A = 512 float4 tasks, B = 256 float4 tasks
  int ra0 = tid >> 2;                  // rows 0..63
  int ra1 = (tid + 256) >> 2;          // rows 64..127
  int ca  = (tid & 3) * 4;             // k-offset 0/4/8/12
  int rb  = tid >> 2;                  // rows 0..63
  const float* Qa0 = Q + (size_t)(m0 + ra0) * D_ + ca;
  const float* Qa1 = Q + (size_t)(m0 + ra1) * D_ + ca;
  const float* Sb  = S + (size_t)(n0 + rb)  * D_ + ca;

  // prologue: stage k0 = 0 into buffer 0
  float4 pa0 = *(const float4*)(Qa0);
  float4 pa1 = *(const float4*)(Qa1);
  float4 pb  = *(const float4*)(Sb);
  {
    float* a0d = &As[0][ra0 * LDA + ca];
    a0d[0] = pa0.x; a0d[1] = pa0.y; a0d[2] = pa0.z; a0d[3] = pa0.w;
    float* a1d = &As[0][ra1 * LDA + ca];
    a1d[0] = pa1.x; a1d[1] = pa1.y; a1d[2] = pa1.z; a1d[3] = pa1.w;
    float* bd = &Bs[0][rb * LDB + ca];
    bd[0] = pb.x; bd[1] = pb.y; bd[2] = pb.z; bd[3] = pb.w;
  }
  __syncthreads();

  v8f acc00 = {}, acc01 = {}, acc10 = {}, acc11 = {};

  for (int k0 = 0; k0 < D_; k0 += KC) {
    int  cur  = (k0 / KC) & 1;
    bool more = (k0 + KC) < D_;
    if (more) {   // prefetch next stage into registers (overlaps with WMMAs)
      pa0 = *(const float4*)(Qa0 + k0 + KC);
      pa1 = *(const float4*)(Qa1 + k0 + KC);
      pb  = *(const float4*)(Sb  + k0 + KC);
    }
    const float* Ap0 = &As[cur][(wm * 32 + lo) * LDA];
    const float* Ap1 = Ap0 + 16 * LDA;
    const float* Bp0 = &Bs[cur][(wn * 32 + lo) * LDB];
    const float* Bp1 = Bp0 + 16 * LDB;
#pragma unroll
    for (int kk = 0; kk < KC; kk += 4) {
      int kb = kk + 2 * hi;
      // A 16x4 f32 frag: lane&15 = M row; VGPR v holds K = 2*(lane>>4)+v
      v2f a0; a0.x = Ap0[kb]; a0.y = Ap0[kb + 1];
      v2f a1; a1.x = Ap1[kb]; a1.y = Ap1[kb + 1];
      // B 4x16 f32 frag: lane&15 = N col; VGPR v holds K = 2*(lane>>4)+v
      v2f b0; b0.x = Bp0[kb]; b0.y = Bp0[kb + 1];
      v2f b1; b1.x = Bp1[kb]; b1.y = Bp1[kb + 1];
      acc00 = WMMA_F32(a0, b0, acc00);
      acc01 = WMMA_F32(a0, b1, acc01);
      acc10 = WMMA_F32(a1, b0, acc10);
      acc11 = WMMA_F32(a1, b1, acc11);
    }
    if (more) {   // store prefetched stage into the other buffer
      int nxt = cur ^ 1;
      float* a0d = &As[nxt][ra0 * LDA + ca];
      a0d[0] = pa0.x; a0d[1] = pa0.y; a0d[2] = pa0.z; a0d[3] = pa0.w;
      float* a1d = &As[nxt][ra1 * LDA + ca];
      a1d[0] = pa1.x; a1d[1] = pa1.y; a1d[2] = pa1.z; a1d[3] = pa1.w;
      float* bd = &Bs[nxt][rb * LDB + ca];
      bd[0] = pb.x; bd[1] = pb.y; bd[2] = pb.z; bd[3] = pb.w;
    }
    __syncthreads();
  }

  // C/D layout: VGPR v, lanes0-15 -> (M=v, N=lane), lanes16-31 -> (M=v+8, N=lane-16)
#pragma unroll
  for (int v = 0; v < 8; ++v) {
    int r0 = wm * 32 + v + 8 * hi;
    int c0 = wn * 32 + lo;
    Ds[r0 * LDD + c0]             = acc00[v];
    Ds[r0 * LDD + c0 + 16]        = acc01[v];
    Ds[(r0 + 16) * LDD + c0]      = acc10[v];
    Ds[(r0 + 16) * LDD + c0 + 16] = acc11[v];
  }
  __syncthreads();

  // per-(query,class) max over this 64-support tile, then one atomic each
  for (int t = tid; t < TM * NCLS_; t += 256) {
    int ql = t / NCLS_;
    int c  = t % NCLS_;
    float q2v  = q2[m0 + ql];
    float best = -__builtin_inff();
    for (int col = 0; col < TN; ++col) {
      if (slab[col] == c) {
        float d = 2.0f * Ds[ql * LDD + col] - q2v - s2t[col];
        best = fmaxf(best, d);
      }
    }
    if (best > -__builtin_inff())
      atomicMaxF(&logits[(size_t)(m0 + ql) * NCLS_ + c], best);
  }
}

// ---------------- argmax (first occurrence, matches jnp.argmax) -------------
__global__ void argmax_kernel(const float* __restrict__ logits,
                              float* __restrict__ pred, int n) {
  int q = blockIdx.x * blockDim.x + threadIdx.x;
  if (q >= n) return;
  float best = logits[(size_t)q * NCLS_];
  int bi = 0;
  for (int c = 1; c < NCLS_; ++c) {
    float v = logits[(size_t)q * NCLS_ + c];
    if (v > best) { best = v; bi = c; }
  }
  pred[q] = (float)bi;
}

extern "C" void kernel_launch(void* const* d_in, const int* in_sizes, int n_in,
                              void* d_out, int out_size, void* d_ws, size_t ws_size,
                              hipStream_t stream) {
  const float* sup_emb  = (const float*)d_in[0];
  const float* qry_emb  = (const float*)d_in[1];
  const int*   sup_mask = (const int*)d_in[2];
  const int*   qry_mask = (const int*)d_in[3];
  const int*   sup_sid  = (const int*)d_in[4];
  const int*   qry_sid  = (const int*)d_in[5];
  const int*   sup_lab  = (const int*)d_in[6];

  char* ws = (char*)d_ws;
  size_t off = 0;
  int*   order_s = (int*)(ws + off);   off += (size_t)B_ * L_ * sizeof(int);
  int*   order_q = (int*)(ws + off);   off += (size_t)B_ * L_ * sizeof(int);
  float* Sspans  = (float*)(ws + off); off += (size_t)NSPAN_ * D_ * sizeof(float);
  float* Qspans  = (float*)(ws + off); off += (size_t)NSPAN_ * D_ * sizeof(float);
  float* s2      = (float*)(ws + off); off += (size_t)NSPAN_ * sizeof(float);
  float* q2      = (float*)(ws + off); off += (size_t)NSPAN_ * sizeof(float);
  (void)ws_size; (void)in_sizes; (void)n_in; (void)out_size;

  order_kernel<<<B_, L_, 0, stream>>>(sup_mask, order_s);
  order_kernel<<<B_, L_, 0, stream>>>(qry_mask, order_q);

  gather_kernel<<<NSPAN_, 256, 0, stream>>>(sup_emb, sup_sid, order_s, Sspans, s2);
  gather_kernel<<<NSPAN_, 256, 0, stream>>>(qry_emb, qry_sid, order_q, Qspans, q2);

  float* logits = (float*)d_out;                  // [4096][6]
  float* pred   = (float*)d_out + NSPAN_ * NCLS_; // [4096]

  init_logits<<<(NSPAN_ * NCLS_ + 255) / 256, 256, 0, stream>>>(logits, NSPAN_ * NCLS_);

  dim3 grid(NSPAN_ / TN, NSPAN_ / TM);            // 64 x 32 blocks
  nn_gemm_kernel<<<grid, 256, 0, stream>>>(Qspans, Sspans, q2, s2, sup_lab, logits);

  argmax_kernel<<<(NSPAN_ + 255) / 256, 256, 0, stream>>>(logits, pred, NSPAN_);
}